// NeuralNetwork_49100066128497
// MI455X (gfx1250) — compile-verified
//
#include <hip/hip_runtime.h>
#include <hip/hip_bf16.h>

// ---------------------------------------------------------------------------
// 64-layer stacked tanh-RNN, wavefront-pipelined persistent kernel.
// One workgroup per layer (64 WGs x 1024 threads). fp32 WMMA 16x16x4.
// Fused GEMMs with manual ping-pong double buffering (no rotation moves).
// Hardware v_tanh_f32 when available.
// ---------------------------------------------------------------------------

#define T_LEN   128
#define BATCH   64
#define HID     512
#define NLAYERS 64
#define HP      516   // LDS row pitch in floats: 516 % 64 == 4 -> conflict-free

typedef __attribute__((ext_vector_type(2))) float v2f;
typedef __attribute__((ext_vector_type(8))) float v8f;

#define WMMA_F32(a, b, c) __builtin_amdgcn_wmma_f32_16x16x4_f32( \
    false, (a), false, (b), (short)0, (c), false, false)

#if __has_builtin(__builtin_amdgcn_tanhf)
  #define TANH(v) __builtin_amdgcn_tanhf(v)   // single V_TANH_F32 trans op
#else
  #define TANH(v) tanhf(v)
#endif

__global__ void rnn_init_flags(unsigned int* flags, int n) {
    int i = blockIdx.x * blockDim.x + threadIdx.x;
    if (i < n) flags[i] = 0u;
}

__global__ void __launch_bounds__(1024, 1)
rnn_stack_kernel(const float* __restrict__ x,
                 const float* __restrict__ Wih_all,
                 const float* __restrict__ Whh_all,
                 const float* __restrict__ bih_all,
                 const float* __restrict__ bhh_all,
                 const float* __restrict__ lin_w,
                 const float* __restrict__ lin_b,
                 float* __restrict__ out,
                 float* __restrict__ buf0,
                 float* __restrict__ buf1,
                 unsigned int* __restrict__ flags) {
    __shared__ float hbuf[BATCH * HP];   // 132,096 B hidden state (fp32)

    const int l    = blockIdx.x;         // layer id
    const int tid  = threadIdx.x;
    const int lane = tid & 31;
    const int wave = tid >> 5;           // 0..31 -> N tile
    const int nl   = lane & 15;          // n (or m) within 16-tile
    const int koff = (lane & 16) ? 2 : 0;// fp32 WMMA A/B frag K sub-offset
    const int n    = wave * 16 + nl;     // global output column

    const float* Wih = Wih_all + (size_t)l * HID * HID;
    const float* Whh = Whh_all + (size_t)l * HID * HID;
    const float  bsum = bih_all[l * HID + n] + bhh_all[l * HID + n];

    // layer 0 reads x, writes buf0; layer 1 reads buf0, writes buf1; ...
    const float* bufIn  = (l == 0) ? x : ((l & 1) ? buf0 : buf1);
    float*       bufOut = (l & 1) ? buf1 : buf0;

    // h_0 = 0
    for (int i = tid; i < BATCH * HP; i += 1024) hbuf[i] = 0.0f;
    __syncthreads();

    const float* wih_row = Wih + (size_t)n * HID + koff; // B frag rows (W[n][k])
    const float* whh_row = Whh + (size_t)n * HID + koff;
    const int    hbase   = nl * HP + koff;               // LDS A frag base
    const int    rbase   = (lane & 16) ? 8 : 0;          // C/D row base per half

    for (int t = 0; t < T_LEN; ++t) {
        // ---- wait for layer l-1 to publish timestep t ----
        if (l > 0) {
            if (tid == 0) {
                const unsigned idx = (unsigned)(l - 1) * T_LEN + t;
                while (__hip_atomic_load(&flags[idx], __ATOMIC_ACQUIRE,
                                         __HIP_MEMORY_SCOPE_AGENT) == 0u) {
                    __builtin_amdgcn_s_sleep(2);
                }
            }
            __syncthreads();
        }

        const float* xin = bufIn + (size_t)t * BATCH * HID;
        const float* xa  = xin + (size_t)nl * HID + koff; // A frag base (global)

        v8f acc[4];
        #pragma unroll
        for (int mt = 0; mt < 4; ++mt) acc[mt] = (v8f)0.0f;

        // ---- fused GEMMs: acc = x_t @ Wih^T + h_{t-1} @ Whh^T ----
        // Ping-pong double buffering over 4-wide k-windows (128 windows).
        auto loadW = [&](int k, v2f& bi, v2f& bh, v2f (&ax)[4], v2f (&ah)[4]) {
            bi = *(const v2f*)(wih_row + k);
            bh = *(const v2f*)(whh_row + k);
            #pragma unroll
            for (int mt = 0; mt < 4; ++mt) {
                ax[mt] = *(const v2f*)(xa + (size_t)(mt * 16) * HID + k);
                ah[mt] = *(const v2f*)(&hbuf[hbase + mt * 16 * HP + k]);
            }
        };
        auto compute = [&](v2f bi, v2f bh, v2f (&ax)[4], v2f (&ah)[4]) {
            #pragma unroll
            for (int mt = 0; mt < 4; ++mt) acc[mt] = WMMA_F32(ax[mt], bi, acc[mt]);
            #pragma unroll
            for (int mt = 0; mt < 4; ++mt) acc[mt] = WMMA_F32(ah[mt], bh, acc[mt]);
        };

        v2f biA, bhA, axA[4], ahA[4];
        v2f biB, bhB, axB[4], ahB[4];
        loadW(0, biA, bhA, axA, ahA);
        #pragma unroll 1
        for (int k = 0; k < HID - 8; k += 8) {
            loadW(k + 4, biB, bhB, axB, ahB);   // prefetch while computing A
            compute(biA, bhA, axA, ahA);
            loadW(k + 8, biA, bhA, axA, ahA);   // prefetch while computing B
            compute(biB, bhB, axB, ahB);
        }
        loadW(HID - 4, biB, bhB, axB, ahB);      // window 508
        compute(biA, bhA, axA, ahA);             // window 504
        compute(biB, bhB, axB, ahB);             // window 508

        __syncthreads();   // everyone done READING old h before overwrite

        if (l < NLAYERS - 1) {
            float* outT = bufOut + (size_t)t * BATCH * HID;
            #pragma unroll
            for (int mt = 0; mt < 4; ++mt) {
                #pragma unroll
                for (int i = 0; i < 8; ++i) {
                    const int m = mt * 16 + rbase + i;
                    float hv = TANH(acc[mt][i] + bsum);
                    hbuf[m * HP + n] = hv;
                    outT[(size_t)m * HID + n] = hv;
                }
            }
            __threadfence();
            __syncthreads();
            if (tid == 0) {
                __hip_atomic_store(&flags[(unsigned)l * T_LEN + t], 1u,
                                   __ATOMIC_RELEASE, __HIP_MEMORY_SCOPE_AGENT);
            }
        } else {
            // top layer: only keep h in LDS (sequence output unused except t=127)
            #pragma unroll
            for (int mt = 0; mt < 4; ++mt) {
                #pragma unroll
                for (int i = 0; i < 8; ++i) {
                    const int m = mt * 16 + rbase + i;
                    hbuf[m * HP + n] = TANH(acc[mt][i] + bsum);
                }
            }
            __syncthreads();
        }
    }

    // ---- linear head on top layer's final hidden state ----
    if (l == NLAYERS - 1) {
        v8f acc[4];
        #pragma unroll
        for (int mt = 0; mt < 4; ++mt) acc[mt] = (v8f)0.0f;

        const float* lw_row = lin_w + (size_t)n * HID + koff;
        #pragma unroll 1
        for (int k = 0; k < HID; k += 4) {
            v2f b = *(const v2f*)(lw_row + k);
            #pragma unroll
            for (int mt = 0; mt < 4; ++mt) {
                v2f a = *(const v2f*)(&hbuf[hbase + mt * 16 * HP + k]);
                acc[mt] = WMMA_F32(a, b, acc[mt]);
            }
        }
        const float lb = lin_b[n];
        #pragma unroll
        for (int mt = 0; mt < 4; ++mt) {
            #pragma unroll
            for (int i = 0; i < 8; ++i) {
                const int m = mt * 16 + rbase + i;
                out[(size_t)m * HID + n] = acc[mt][i] + lb;
            }
        }
    }
}

extern "C" void kernel_launch(void* const* d_in, const int* in_sizes, int n_in,
                              void* d_out, int out_size, void* d_ws, size_t ws_size,
                              hipStream_t stream) {
    const float* x     = (const float*)d_in[0];  // (T,B,H)
    const float* W_ih  = (const float*)d_in[1];  // (L,H,H)
    const float* W_hh  = (const float*)d_in[2];  // (L,H,H)
    const float* b_ih  = (const float*)d_in[3];  // (L,H)
    const float* b_hh  = (const float*)d_in[4];  // (L,H)
    const float* lin_w = (const float*)d_in[5];  // (H,H)
    const float* lin_b = (const float*)d_in[6];  // (H,)
    float* out = (float*)d_out;                  // (B,H)

    // workspace layout: [flags: 64*128 u32 = 32KB][buf0: 16MB][buf1: 16MB]
    unsigned int* flags = (unsigned int*)d_ws;
    float* buf0 = (float*)((char*)d_ws + 32768);
    float* buf1 = buf0 + (size_t)T_LEN * BATCH * HID;

    const int nflags = NLAYERS * T_LEN;
    rnn_init_flags<<<(nflags + 255) / 256, 256, 0, stream>>>(flags, nflags);

    rnn_stack_kernel<<<NLAYERS, 1024, 0, stream>>>(
        x, W_ih, W_hh, b_ih, b_hh, lin_w, lin_b, out, buf0, buf1, flags);
}